// ObjectCondensation_18708877541911
// MI455X (gfx1250) — compile-verified
//
#include <hip/hip_runtime.h>
#include <stdint.h>

#define N_HITS   100000
#define D_EMB    12
#define N_OBJ    1024
#define Q_MIN    0.1f
#define SB_W     0.1f
#define EPS_F    1e-9f
#define MAX_NREP 100000.0f

typedef float v2f __attribute__((ext_vector_type(2)));
typedef float v8f __attribute__((ext_vector_type(8)));
typedef unsigned int u32x4 __attribute__((ext_vector_type(4)));
typedef int i32x4 __attribute__((ext_vector_type(4)));
typedef int i32x8 __attribute__((ext_vector_type(8)));

#define NW_TILES ((N_HITS + 15) / 16)          // 6250 row tiles (1 wave each)
#define WAVES_PER_BLOCK 8
#define NB_PAIR  ((NW_TILES + WAVES_PER_BLOCK - 1) / WAVES_PER_BLOCK)
#define NB_HIT   ((N_HITS + 255) / 256)        // 391
#define NB_OBJ   ((N_OBJ + 255) / 256)         // 4
#define STAGE_F  (N_OBJ * D_EMB + 2 * N_OBJ)   // XA | QAA(interleaved), 14336 f = 57344 B

// ---------------------------------------------------------------- kernel 0
__global__ void oc_init_kernel(unsigned long long* keys) {
    int k = blockIdx.x * blockDim.x + threadIdx.x;
    if (k < N_OBJ) keys[k] = 0ull;
}

// ---------------------------------------------------------------- kernel 1: per hit
__global__ __launch_bounds__(256)
void oc_hit_kernel(const float* __restrict__ beta, const float* __restrict__ x,
                   const int* __restrict__ oid,
                   float* __restrict__ q, float* __restrict__ xx,
                   unsigned long long* __restrict__ keys,
                   float* __restrict__ noiseS, unsigned* __restrict__ noiseC) {
    int j = blockIdx.x * blockDim.x + threadIdx.x;
    float b = 0.f, qq = 0.f;
    int id = -1;
    if (j < N_HITS) {
        b  = beta[j];
        id = oid[j];
        float a = atanhf(b);
        qq = a * a + Q_MIN;                    // qq > 0 -> float bits orderable as uint
        q[j] = qq;
        float s = 0.f;
        const float* xr = x + (size_t)j * D_EMB;
        #pragma unroll
        for (int d = 0; d < D_EMB; ++d) { float v = xr[d]; s += v * v; }
        xx[j] = s;
        if (id >= 1 && id <= N_OBJ) {
            // argmax with first-index tie-break: bigger q wins, then smaller j wins
            unsigned long long key =
                ((unsigned long long)__float_as_uint(qq) << 32) |
                (unsigned long long)(0xFFFFFFFFu - (unsigned)j);
            atomicMax(&keys[id - 1], key);
        }
    }
    float    ns = (j < N_HITS && id <= 0) ? b  : 0.f;
    unsigned nc = (j < N_HITS && id <= 0) ? 1u : 0u;
    __shared__ float    sf[256];
    __shared__ unsigned su[256];
    int tid = threadIdx.x;
    sf[tid] = ns; su[tid] = nc;
    __syncthreads();
    for (int s = 128; s > 0; s >>= 1) {
        if (tid < s) { sf[tid] += sf[tid + s]; su[tid] += su[tid + s]; }
        __syncthreads();
    }
    if (tid == 0) { noiseS[blockIdx.x] = sf[0]; noiseC[blockIdx.x] = su[0]; }
}

// ---------------------------------------------------------------- kernel 2: per object
// stage = [ XA (1024x12) | QAA (1024 x {q_a, |x_a|^2}) ] contiguous in ws
__global__ __launch_bounds__(256)
void oc_obj_kernel(const float* __restrict__ beta, const float* __restrict__ x,
                   const float* __restrict__ q, const float* __restrict__ xx,
                   const unsigned long long* __restrict__ keys,
                   float* __restrict__ stage, float* __restrict__ cowP) {
    float* XA  = stage;
    float* QAA = stage + N_OBJ * D_EMB;
    int k = blockIdx.x * blockDim.x + threadIdx.x;
    float cw = 0.f;
    if (k < N_OBJ) {
        unsigned long long key = keys[k];
        // empty object column -> argmax of zeros -> index 0 (matches jnp.argmax)
        unsigned alpha = (key == 0ull) ? 0u
                       : (0xFFFFFFFFu - (unsigned)(key & 0xFFFFFFFFull));
        QAA[2 * k + 0] = q[alpha];
        QAA[2 * k + 1] = xx[alpha];
        const float* xr = x + (size_t)alpha * D_EMB;
        float* xa = XA + (size_t)k * D_EMB;
        #pragma unroll
        for (int d = 0; d < D_EMB; ++d) xa[d] = xr[d];
        cw = 1.0f - beta[alpha];
    }
    __shared__ float sf[256];
    int tid = threadIdx.x;
    sf[tid] = cw;
    __syncthreads();
    for (int s = 128; s > 0; s >>= 1) {
        if (tid < s) sf[tid] += sf[tid + s];
        __syncthreads();
    }
    if (tid == 0) cowP[blockIdx.x] = sf[0];
}

// ---------------------------------------------------------------- kernel 3: fused WMMA pair pass
// One wave owns a 16-hit row tile, sweeps 64 object column tiles.
// dot() tile = 3 chained V_WMMA_F32_16X16X4_F32 (d=12 = 3 K-slabs of 4).
// Branch-free epilogue: v_cmp/v_cndmask selects + raw v_sqrt_f32 (TRANS).
// Condensation-point data staged to LDS once per block via the Tensor Data
// Mover. smem stays in the LDS address space (ds_load_b64 in the loop); an
// opaque never-taken store defeats "no stores -> loads fold to undef".
__global__ __launch_bounds__(256)
void oc_pair_kernel(const float* __restrict__ x, const int* __restrict__ oid,
                    const float* __restrict__ q, const float* __restrict__ xx,
                    const float* __restrict__ stage,
                    float* __restrict__ attP, float* __restrict__ repP,
                    unsigned* __restrict__ cntP) {
    __shared__ float smem[STAGE_F];            // sole LDS object -> LDS offset 0

#if defined(__gfx1250__) && __has_builtin(__builtin_amdgcn_tensor_load_to_lds) && \
    __has_builtin(__builtin_amdgcn_s_wait_tensorcnt)
    // Tensor Data Mover: one wave DMAs the whole 57,344 B stage into LDS.
    // 1-D D#: count=1, data_size=4B, tile_dim0=tensor_dim0=stride=14336 elems.
    if (threadIdx.x < 32) {
        uint64_t ga = (uint64_t)(uintptr_t)stage;
        u32x4 g0;
        g0.x = 1u;                                           // count=1 (valid D#)
        g0.y = 0u;                                           // lds_addr = 0
        g0.z = (unsigned)(ga & 0xFFFFFFFFull);               // global_addr[31:0]
        g0.w = (unsigned)((ga >> 32) & 0x01FFFFFFull)        // global_addr[56:32]
             | (2u << 30);                                   // type=2 ("image")
        i32x8 g1 = (i32x8)0;
        g1[0] = (int)(2u << 16);                             // data_size=4B
        g1[1] = (int)((unsigned)STAGE_F << 16);              // tensor_dim0[15:0] @63:48
        g1[2] = (int)(1u << 16);                             // tensor_dim1=1 @111:80
        g1[3] = (int)((unsigned)STAGE_F << 16);              // tile_dim0 @127:112
        g1[5] = STAGE_F;                                     // tensor_dim0_stride lo32
        i32x4 gz4 = (i32x4)0;
        i32x8 gz8 = (i32x8)0;
        __builtin_amdgcn_tensor_load_to_lds(g0, g1, gz4, gz4, gz8, 0);
        __builtin_amdgcn_s_wait_tensorcnt(0);
    }
    __syncthreads();
#else
    for (int i = threadIdx.x; i < STAGE_F / 4; i += 256)
        ((float4*)smem)[i] = ((const float4*)stage)[i];
    __syncthreads();
#endif

    // Opaque never-taken store: the TDM's LDS writes are invisible to the
    // optimizer, and with zero visible stores it folds all smem loads to
    // undef and deletes the WMMA loop (round-3 regression). The asm output
    // is 0 at runtime but unknowable at compile time, so the store below
    // must be assumed possible -- keeping smem "stored" while leaving the
    // pointer in the LDS address space (ds_load, not flat).
    int never;
    asm volatile("" : "=s"(never) : "0"(0));
    if (never) smem[0] = 0.0f;

    const float* sXA  = smem;
    const float* sQAA = smem + N_OBJ * D_EMB;

    const int lane   = threadIdx.x & 31;
    const int waveId = blockIdx.x * WAVES_PER_BLOCK + (threadIdx.x >> 5);
    if (waveId >= NW_TILES) return;            // wave-uniform: EXEC all-1 for WMMA

    const int j0    = waveId * 16;
    const int ln    = lane & 15;
    const int half  = lane >> 4;               // wave32 A/B layout: lanes 0-15 / 16-31
    const int dbase = 2 * half;

    // A tiles (16x4 f32, 2 VGPRs each): lane holds row m=ln, K-slice by half
    v2f a0, a1, a2;
    {
        int j = j0 + ln;
        bool ok = j < N_HITS;
        const float* xr = x + (size_t)j * D_EMB;
        a0.x = ok ? xr[0 + dbase] : 0.f;  a0.y = ok ? xr[1 + dbase] : 0.f;
        a1.x = ok ? xr[4 + dbase] : 0.f;  a1.y = ok ? xr[5 + dbase] : 0.f;
        a2.x = ok ? xr[8 + dbase] : 0.f;  a2.y = ok ? xr[9 + dbase] : 0.f;
    }
    // Per-row state for the 8 D-matrix rows this lane owns (m = v + 8*half)
    float qrow[8], xxrow[8];
    int   orow[8];
    #pragma unroll
    for (int v = 0; v < 8; ++v) {
        int j = j0 + v + 8 * half;
        bool ok = j < N_HITS;
        qrow[v]  = ok ? q[j]   : 0.f;
        xxrow[v] = ok ? xx[j]  : 0.f;
        orow[v]  = ok ? oid[j] : -1;           // -1 can never equal an object id
    }

    float attacc = 0.f, repacc = 0.f;
    unsigned cnt = 0;

    for (int c = 0; c < N_OBJ / 16; ++c) {
        const int kcol = c * 16 + ln;          // this lane's output column
        const float* br = sXA + (size_t)kcol * D_EMB;
        v2f b0 = *(const v2f*)(br + 0 + dbase);        // ds_load_b64 x3
        v2f b1 = *(const v2f*)(br + 4 + dbase);
        v2f b2 = *(const v2f*)(br + 8 + dbase);
        v2f qa_aa = *(const v2f*)(sQAA + 2 * kcol);    // ds_load_b64 {q_a, |x_a|^2}
        const float qa   = qa_aa.x;
        const float aa   = qa_aa.y;
        const int   oidc = kcol + 1;

        v8f acc = {};
        acc = __builtin_amdgcn_wmma_f32_16x16x4_f32(false, a0, false, b0,
                                                    (short)0, acc, false, false);
        acc = __builtin_amdgcn_wmma_f32_16x16x4_f32(false, a1, false, b1,
                                                    (short)0, acc, false, false);
        acc = __builtin_amdgcn_wmma_f32_16x16x4_f32(false, a2, false, b2,
                                                    (short)0, acc, false, false);

        #pragma unroll
        for (int v = 0; v < 8; ++v) {          // D[v] = row (v+8*half), col ln
            float d2   = fmaxf(xxrow[v] + aa - 2.0f * acc[v], 0.f);
            float dist = __builtin_amdgcn_sqrtf(fmaxf(d2, 1e-12f));
            float qw   = qrow[v] * qa;
            bool  att  = (orow[v] == oidc);            // false for padded rows
            bool  rep  = !att && (dist < 1.0f) && (orow[v] >= 0);
            attacc += att ? qw * d2 : 0.f;             // attractive (v_cndmask+v_add)
            repacc += rep ? qw * (1.0f - dist) : 0.f;  // repulsive hinge
            cnt    += rep ? 1u : 0u;
        }
    }

    // wave32 butterfly reduction, then one deterministic slot per wave
    #pragma unroll
    for (int off = 16; off > 0; off >>= 1) {
        attacc += __shfl_xor(attacc, off, 32);
        repacc += __shfl_xor(repacc, off, 32);
        cnt    += __shfl_xor(cnt,    off, 32);
    }
    if (lane == 0) { attP[waveId] = attacc; repP[waveId] = repacc; cntP[waveId] = cnt; }
}

// ---------------------------------------------------------------- kernel 4: finalize
__global__ __launch_bounds__(256)
void oc_final_kernel(const float* __restrict__ attP, const float* __restrict__ repP,
                     const unsigned* __restrict__ cntP,
                     const float* __restrict__ noiseS, const unsigned* __restrict__ noiseC,
                     const float* __restrict__ cowP, float* __restrict__ out) {
    __shared__ float    sf[256];
    __shared__ unsigned su[256];
    int tid = threadIdx.x;

    float as = 0.f, rs = 0.f, ns = 0.f, cw = 0.f;
    unsigned cs = 0, nc = 0;
    for (int i = tid; i < NW_TILES; i += 256) { as += attP[i]; rs += repP[i]; cs += cntP[i]; }
    for (int i = tid; i < NB_HIT;   i += 256) { ns += noiseS[i]; nc += noiseC[i]; }
    for (int i = tid; i < NB_OBJ;   i += 256) { cw += cowP[i]; }

    float    fvals[4] = {as, rs, ns, cw};
    unsigned uvals[2] = {cs, nc};
    #pragma unroll
    for (int r = 0; r < 4; ++r) {
        sf[tid] = fvals[r];
        __syncthreads();
        for (int s = 128; s > 0; s >>= 1) {
            if (tid < s) sf[tid] += sf[tid + s];
            __syncthreads();
        }
        if (tid == 0) fvals[r] = sf[0];
        __syncthreads();
    }
    #pragma unroll
    for (int r = 0; r < 2; ++r) {
        su[tid] = uvals[r];
        __syncthreads();
        for (int s = 128; s > 0; s >>= 1) {
            if (tid < s) su[tid] += su[tid + s];
            __syncthreads();
        }
        if (tid == 0) uvals[r] = su[0];
        __syncthreads();
    }

    if (tid == 0) {
        float att_total = fvals[0], rep_total = fvals[1];
        float noise_sum = fvals[2], cow_total = fvals[3];
        unsigned n_rep = uvals[0], n_noise = uvals[1];

        float v_att = att_total / (float)N_HITS;
        // Reference subsamples rep pairs at rate freq and upweights by 1/freq;
        // the fused full sum equals that estimator's expectation (and is exact
        // whenever n_rep <= MAX_NREP, since then freq == 1, samp all-true).
        float freq = fminf(MAX_NREP / fmaxf((float)n_rep, 1.0f), 1.0f);
        (void)freq;
        float v_rep = rep_total / (float)N_HITS;
        float l_cow = cow_total / (float)N_OBJ;
        float l_noise = noise_sum / (fmaxf((float)n_noise, 1.0f) + EPS_F);

        out[0] = v_att;
        out[1] = v_rep;
        out[2] = l_cow;
        out[3] = l_noise;
        out[4] = v_att + v_rep + l_cow + l_noise * SB_W;
    }
}

// ---------------------------------------------------------------- launcher
extern "C" void kernel_launch(void* const* d_in, const int* in_sizes, int n_in,
                              void* d_out, int out_size, void* d_ws, size_t ws_size,
                              hipStream_t stream) {
    const float* beta = (const float*)d_in[0];
    const float* x    = (const float*)d_in[1];
    const int*   oid  = (const int*)d_in[2];
    float*       out  = (float*)d_out;

    // workspace carve-up (16-byte aligned slabs; XA|QAA kept contiguous)
    char* ws = (char*)d_ws;
    size_t off = 0;
    auto take = [&](size_t bytes) -> char* {
        off = (off + 15) & ~(size_t)15;
        char* p = ws + off;
        off += bytes;
        return p;
    };
    float*              q      = (float*)take(sizeof(float) * N_HITS);
    float*              xx     = (float*)take(sizeof(float) * N_HITS);
    unsigned long long* keys   = (unsigned long long*)take(sizeof(unsigned long long) * N_OBJ);
    float*              stage  = (float*)take(sizeof(float) * STAGE_F);   // XA|QAA
    float*              attP   = (float*)take(sizeof(float) * NW_TILES);
    float*              repP   = (float*)take(sizeof(float) * NW_TILES);
    unsigned*           cntP   = (unsigned*)take(sizeof(unsigned) * NW_TILES);
    float*              noiseS = (float*)take(sizeof(float) * NB_HIT);
    unsigned*           noiseC = (unsigned*)take(sizeof(unsigned) * NB_HIT);
    float*              cowP   = (float*)take(sizeof(float) * NB_OBJ);
    (void)ws_size; (void)in_sizes; (void)n_in; (void)out_size;

    oc_init_kernel<<<(N_OBJ + 255) / 256, 256, 0, stream>>>(keys);
    oc_hit_kernel<<<NB_HIT, 256, 0, stream>>>(beta, x, oid, q, xx, keys, noiseS, noiseC);
    oc_obj_kernel<<<NB_OBJ, 256, 0, stream>>>(beta, x, q, xx, keys, stage, cowP);
    oc_pair_kernel<<<NB_PAIR, 256, 0, stream>>>(x, oid, q, xx, stage, attP, repP, cntP);
    oc_final_kernel<<<1, 256, 0, stream>>>(attP, repP, cntP, noiseS, noiseC, cowP, out);
}